// ContrastiveTopK_712964571591
// MI455X (gfx1250) — compile-verified
//
#include <hip/hip_runtime.h>
#include <stdint.h>

#define VOCAB    50257
#define KKEEP    45232          /* ceil(0.9 * 50257) */
#define TOPK     10
#define NTHREADS 512            /* 16 wave32 waves */
#define EPS      1e-8f

/* ---------------- dynamic LDS layout (bytes) ----------------
   keys: full row (50257 u32)  -> the 320KB WGP LDS is the whole point  */
#define KEYS_OFF   0
#define KEYS_BYTES 201056                      /* 50257*4 rounded to 32 */
#define HIST_OFF   (KEYS_OFF + KEYS_BYTES)     /* 2048 u32 */
#define COARSE_OFF (HIST_OFF + 2048 * 4)       /* 512 u32  */
#define WRED_OFF   (COARSE_OFF + 512 * 4)      /* 32 f32   */
#define BC_OFF     (WRED_OFF + 32 * 4)         /* 8 u32    */
#define CIDX_OFF   (BC_OFF + 8 * 4)            /* 64 u32   */
#define CKEY_OFF   (CIDX_OFF + 64 * 4)         /* 64 u32   */
#define FIDX_OFF   (CKEY_OFF + 64 * 4)         /* 16 u32   */
#define FVAL_OFF   (FIDX_OFF + 16 * 4)         /* 16 f32   */
#define FSC_OFF    (FVAL_OFF + 16 * 4)         /* 16 f32   */
#define CNT_OFF    (FSC_OFF + 16 * 4)          /* 1 u32    */
#define SMEM_BYTES (CNT_OFF + 64)              /* ~212 KB < 320 KB */

typedef unsigned int t_v4u __attribute__((ext_vector_type(4)));
typedef int          t_v4i __attribute__((ext_vector_type(4)));
typedef int          t_v8i __attribute__((ext_vector_type(8)));
typedef float        w_v2f __attribute__((ext_vector_type(2)));
typedef float        w_v8f __attribute__((ext_vector_type(8)));

/* monotone float<->uint key transform (greater key <=> greater float) */
__device__ __forceinline__ unsigned f2key(float f) {
    unsigned u = __float_as_uint(f);
    return (u & 0x80000000u) ? ~u : (u | 0x80000000u);
}
__device__ __forceinline__ float key2f(unsigned k) {
    unsigned u = (k & 0x80000000u) ? (k & 0x7FFFFFFFu) : ~k;
    return __uint_as_float(u);
}

/* ---------- TDM: DMA one contiguous row (n f32) from global into LDS ----------
   One descriptor, issued by wave 0 only (TENSORcnt is per-wave); other waves
   wait at the workgroup barrier placed by the caller. */
__device__ __forceinline__ void tdm_load_row(const float* gsrc, unsigned* dst, int n) {
#if __has_builtin(__builtin_amdgcn_tensor_load_to_lds)
    if (threadIdx.x < 32) {
        unsigned long long ga = (unsigned long long)(uintptr_t)gsrc;
        unsigned lds_off = (unsigned)(unsigned long long)(uintptr_t)dst;
        t_v4u g0;
        g0.x = 1u;                                   /* count=1, user mode */
        g0.y = lds_off;                              /* lds_addr [63:32] */
        g0.z = (unsigned)(ga & 0xFFFFFFFFu);         /* global_addr lo */
        g0.w = (unsigned)((ga >> 32) & 0x01FFFFFFu)  /* global_addr hi (57b) */
             | (2u << 30);                           /* type=2 ("image") */
        t_v8i g1;
        g1[0] = (int)(2u << 16);                          /* data_size=4B */
        g1[1] = (int)(((unsigned)n & 0xFFFFu) << 16);     /* tensor_dim0 lo16 */
        g1[2] = (int)(((unsigned)n >> 16) | (1u << 16));  /* dim0 hi | tensor_dim1=1 */
        g1[3] = (int)(((unsigned)n & 0xFFFFu) << 16);     /* tile_dim0 = n */
        g1[4] = 0;                                        /* tile_dim1/2 unused */
        g1[5] = (int)(unsigned)n;                         /* tensor_dim0_stride lo */
        g1[6] = 0;
        g1[7] = 0;
        t_v4i z4 = {0, 0, 0, 0};
#if defined(__clang_major__) && __clang_major__ >= 23
        t_v8i z8 = {0, 0, 0, 0, 0, 0, 0, 0};
        __builtin_amdgcn_tensor_load_to_lds(g0, g1, z4, z4, z8, 0);
#else
        __builtin_amdgcn_tensor_load_to_lds(g0, g1, z4, z4, 0);
#endif
        __builtin_amdgcn_s_wait_tensorcnt(0);
    }
#else
    for (int i = threadIdx.x; i < n; i += NTHREADS)
        dst[i] = ((const unsigned*)gsrc)[i];
#endif
}

/* ---------- wave32 full-EXEC sum reduction through the matrix pipe ----------
   A(16x4 f32): lane m holds A[m][0]=p (lanes 0-15) / A[m][2]=p (lanes 16-31),
   other K slots 0. B = ones. D[m][*] = p_m + p_{m+16}. Each lane sums its 8
   D VGPRs -> half-wave sums in lanes 0-15 / 16-31; combine with two shfl. */
__device__ __forceinline__ float wave_sum_wmma(float p) {
#if __has_builtin(__builtin_amdgcn_wmma_f32_16x16x4_f32)
    w_v2f a; a.x = p;   a.y = 0.f;
    w_v2f b; b.x = 1.f; b.y = 1.f;
    w_v8f c = {0.f, 0.f, 0.f, 0.f, 0.f, 0.f, 0.f, 0.f};
    w_v8f d = __builtin_amdgcn_wmma_f32_16x16x4_f32(false, a, false, b,
                                                    (short)0, c, false, false);
    float t = d[0] + d[1] + d[2] + d[3] + d[4] + d[5] + d[6] + d[7];
    return __shfl(t, 0, 32) + __shfl(t, 16, 32);
#else
    for (int o = 16; o > 0; o >>= 1) p += __shfl_xor(p, o, 32);
    return p;
#endif
}

__device__ __forceinline__ float block_sum(float v, float* wred) {
    for (int o = 16; o > 0; o >>= 1) v += __shfl_xor(v, o, 32);
    int w = threadIdx.x >> 5;
    if ((threadIdx.x & 31) == 0) wred[w] = v;
    __syncthreads();
    if (threadIdx.x < 32) {
        float p = (threadIdx.x < (NTHREADS >> 5)) ? wred[threadIdx.x] : 0.f;
        float r = wave_sum_wmma(p);
        if (threadIdx.x == 0) wred[0] = r;
    }
    __syncthreads();
    float r = wred[0];
    __syncthreads();
    return r;
}

__device__ __forceinline__ float block_max(float v, float* wred) {
    for (int o = 16; o > 0; o >>= 1) v = fmaxf(v, __shfl_xor(v, o, 32));
    int w = threadIdx.x >> 5;
    if ((threadIdx.x & 31) == 0) wred[w] = v;
    __syncthreads();
    if (threadIdx.x < 32) {
        float p = (threadIdx.x < (NTHREADS >> 5)) ? wred[threadIdx.x]
                                                  : __int_as_float(0xFF800000);
        for (int o = 16; o > 0; o >>= 1) p = fmaxf(p, __shfl_xor(p, o, 32));
        if (threadIdx.x == 0) wred[0] = p;
    }
    __syncthreads();
    float r = wred[0];
    __syncthreads();
    return r;
}

/* ---- one radix level: histogram + block-wide scan -> (bin, residual rank) ---- */
__device__ void level_scan(unsigned* hist, int nbins, unsigned rank,
                           unsigned* coarse, unsigned* bc) {
    const int tid = threadIdx.x;
    const int g = nbins >> 9;                  /* bins per thread (4 or 2) */
    unsigned s = 0;
    for (int j = 0; j < g; ++j) s += hist[tid * g + j];
    coarse[tid] = s;
    __syncthreads();
    for (int off = 1; off < NTHREADS; off <<= 1) {   /* inclusive scan */
        unsigned add = (tid >= off) ? coarse[tid - off] : 0u;
        __syncthreads();
        coarse[tid] += add;
        __syncthreads();
    }
    unsigned total = coarse[NTHREADS - 1];
    unsigned Q     = total - rank;             /* rank in [1, total]        */
    unsigned pincl = coarse[tid];
    unsigned pexcl = (tid == 0) ? 0u : coarse[tid - 1];
    if (pexcl <= Q && Q < pincl) {             /* unique boundary group     */
        unsigned cum = total - pincl;          /* strictly above this group */
        for (int j = g - 1; j >= 0; --j) {
            unsigned h = hist[tid * g + j];
            cum += h;
            if (cum >= rank) {
                bc[0] = (unsigned)(tid * g + j);
                bc[1] = rank - (cum - h);      /* residual within bin       */
                break;
            }
        }
    }
    __syncthreads();
}

/* exact 32-bit select of the `rank`-th largest key: 11+11+10 bit levels.
   Returns T (key value at that rank) and cnt_gt (# keys strictly > T). */
__device__ void radix_select(const unsigned* keys, int n, unsigned rank,
                             unsigned* hist, unsigned* coarse, unsigned* bc,
                             unsigned* outT, unsigned* outCntGt) {
    const int shifts[3] = {21, 10, 0};
    const int nbin[3]   = {2048, 2048, 1024};
    unsigned prefix = 0, pmask = 0, r = rank;
    for (int L = 0; L < 3; ++L) {
        for (int i = threadIdx.x; i < nbin[L]; i += NTHREADS) hist[i] = 0;
        __syncthreads();
        for (int i = threadIdx.x; i < n; i += NTHREADS) {
            unsigned k = keys[i];
            if ((k & pmask) == prefix)
                atomicAdd(&hist[(k >> shifts[L]) & (unsigned)(nbin[L] - 1)], 1u);
        }
        __syncthreads();
        level_scan(hist, nbin[L], r, coarse, bc);
        unsigned b = bc[0];
        r = bc[1];
        prefix |= b << shifts[L];
        pmask  |= (unsigned)(nbin[L] - 1) << shifts[L];
    }
    *outT = prefix;
    *outCntGt = rank - r;
}

__global__ void __launch_bounds__(NTHREADS)
ContrastiveTopK_kernel(const float* __restrict__ lexp,
                       const float* __restrict__ lama,
                       float* __restrict__ out) {
    extern __shared__ char smem[];
    unsigned* keys   = (unsigned*)(smem + KEYS_OFF);
    unsigned* hist   = (unsigned*)(smem + HIST_OFF);
    unsigned* coarse = (unsigned*)(smem + COARSE_OFF);
    float*    wred   = (float*)(smem + WRED_OFF);
    unsigned* bc     = (unsigned*)(smem + BC_OFF);
    unsigned* cIdx   = (unsigned*)(smem + CIDX_OFF);
    unsigned* cKey   = (unsigned*)(smem + CKEY_OFF);
    unsigned* fIdx   = (unsigned*)(smem + FIDX_OFF);
    float*    fVal   = (float*)(smem + FVAL_OFF);
    float*    fSc    = (float*)(smem + FSC_OFF);
    unsigned* cnt    = (unsigned*)(smem + CNT_OFF);

    const int tid = threadIdx.x;
    const int row = blockIdx.x;
    const size_t base = (size_t)row * VOCAB;

    /* ================= phase E: expert row ================= */
    tdm_load_row(lexp + base, keys, VOCAB);
    __syncthreads();

    float mloc = __int_as_float(0xFF800000);
    for (int i = tid; i < VOCAB; i += NTHREADS) {
        float f = __uint_as_float(keys[i]);
        mloc = fmaxf(mloc, f);
        keys[i] = f2key(f);
    }
    __syncthreads();
    const float m_exp = block_max(mloc, wred);

    unsigned Tkeep, cgKeep, T10, cg10;
    radix_select(keys, VOCAB, KKEEP, hist, coarse, bc, &Tkeep, &cgKeep);
    radix_select(keys, VOCAB, TOPK,  hist, coarse, bc, &T10,   &cg10);

    if (tid == 0) *cnt = 0;
    __syncthreads();

    /* denominator sum over the kept set + top-K candidate collection */
    float acc = 0.f;
    for (int i = tid; i < VOCAB; i += NTHREADS) {
        unsigned k = keys[i];
        if (k > Tkeep) acc += expf(key2f(k) - m_exp);
        if (k >= T10) {
            unsigned s = atomicAdd(cnt, 1u);
            if (s < 64u) { cIdx[s] = (unsigned)i; cKey[s] = k; }
        }
    }
    if (tid == 0)  /* ties at the keep-threshold contribute (KKEEP - cnt_gt) copies */
        acc += (float)(KKEEP - cgKeep) * expf(key2f(Tkeep) - m_exp);
    const float S_exp = block_sum(acc, wred);
    const float logS_exp = logf(S_exp);

    /* finalize the <=10 mask positions (tie-break at T10 by lowest index) */
    if (tid == 0) {
        int n = (int)((*cnt > 64u) ? 64u : *cnt);
        int ne_allowed = TOPK - (int)cg10;
        int nf = 0;
        for (int i = 0; i < n && nf < TOPK; ++i)
            if (cKey[i] > T10) { fIdx[nf] = cIdx[i]; fVal[nf] = key2f(cKey[i]); ++nf; }
        for (int pick = 0; pick < ne_allowed && nf < TOPK; ++pick) {
            unsigned best = 0xFFFFFFFFu; int bj = -1;
            for (int i = 0; i < n; ++i)
                if (cKey[i] == T10 && cIdx[i] < best) { best = cIdx[i]; bj = i; }
            if (bj < 0) break;
            cKey[bj] = 0u;
            fIdx[nf] = best; fVal[nf] = key2f(T10); ++nf;
        }
        bc[4] = (unsigned)nf;
    }
    __syncthreads();
    const int nf = (int)bc[4];
    __syncthreads();

    /* ================= phase A: amateur row ================= */
    tdm_load_row(lama + base, keys, VOCAB);
    __syncthreads();

    mloc = __int_as_float(0xFF800000);
    for (int i = tid; i < VOCAB; i += NTHREADS) {
        float f = __uint_as_float(keys[i]);
        mloc = fmaxf(mloc, f);
        keys[i] = f2key(f);
    }
    __syncthreads();
    const float m_ama = block_max(mloc, wred);

    unsigned Tama, cgAma;
    radix_select(keys, VOCAB, KKEEP, hist, coarse, bc, &Tama, &cgAma);

    acc = 0.f;
    for (int i = tid; i < VOCAB; i += NTHREADS) {
        unsigned k = keys[i];
        if (k > Tama) acc += expf(key2f(k) - m_ama);
    }
    if (tid == 0)
        acc += (float)(KKEEP - cgAma) * expf(key2f(Tama) - m_ama);
    const float S_ama = block_sum(acc, wred);

    /* scores at the masked positions */
    if (tid < nf) {
        unsigned ka = keys[fIdx[tid]];
        bool kept = (ka >= Tama);
        float pa = kept ? expf(key2f(ka) - m_ama) / S_ama : 0.f;
        float log_pe = (fVal[tid] - m_exp) - logS_exp;
        fSc[tid] = log_pe - logf(pa + EPS);
    }
    __syncthreads();

    /* ================= output: -inf fill + <=10 fixups ================= */
    float* orow = out + base;
    const float NEGINF = __int_as_float(0xFF800000);
    for (int i = tid; i < VOCAB; i += NTHREADS) orow[i] = NEGINF;
    __threadfence();
    __syncthreads();
    if (tid < nf) orow[fIdx[tid]] = fSc[tid];
}

extern "C" void kernel_launch(void* const* d_in, const int* in_sizes, int n_in,
                              void* d_out, int out_size, void* d_ws, size_t ws_size,
                              hipStream_t stream) {
    (void)n_in; (void)d_ws; (void)ws_size; (void)out_size;
    const float* lexp = (const float*)d_in[0];
    const float* lama = (const float*)d_in[1];
    float* out = (float*)d_out;
    int rows = in_sizes[0] / VOCAB;
    if (rows <= 0) return;
    hipFuncSetAttribute((const void*)ContrastiveTopK_kernel,
                        hipFuncAttributeMaxDynamicSharedMemorySize,
                        (int)SMEM_BYTES);
    ContrastiveTopK_kernel<<<rows, NTHREADS, SMEM_BYTES, stream>>>(lexp, lama, out);
}